// CodeBook_678604833408
// MI455X (gfx1250) — compile-verified
//
#include <hip/hip_runtime.h>
#include <hip/hip_bf16.h>

#define B_SZ 32768
#define K_SZ 4096
#define D_SZ 256

#define SLAB 16          // codebook rows per LDS slab (one 16-wide N tile)
#define LDS_STRIDE 260   // floats per code row in LDS; pad 4 -> (4n+k) bank pattern, conflict-free b64 reads
#define NSTEP (D_SZ / 4) // 64 WMMA K-steps per tile

typedef __attribute__((ext_vector_type(2))) float v2f;
typedef __attribute__((ext_vector_type(4))) float v4f;
typedef __attribute__((ext_vector_type(8))) float v8f;

// ---------------- Kernel 1: e_sq[k] = sum_d w[k][d]^2 (one wave per row) ----------------
__global__ void esq_kernel(const float* __restrict__ w, float* __restrict__ esq) {
  const int lane = threadIdx.x & 31;
  const int wave = threadIdx.x >> 5;
  const int row  = blockIdx.x * 8 + wave;
  const float* p = w + (size_t)row * D_SZ;
  float s = 0.f;
#pragma unroll
  for (int i = 0; i < D_SZ / 32; ++i) {
    float x = p[lane + i * 32];
    s += x * x;
  }
#pragma unroll
  for (int m = 16; m >= 1; m >>= 1) s += __shfl_xor(s, m, 32);
  if (lane == 0) esq[row] = s;
}

// ---------------- Kernel 2: fused GEMM (f32 WMMA) + running argmin ----------------
// Block = 256 threads = 8 waves; each wave owns a 16-row M tile (block covers 128 rows).
// A (16x256 z tile) lives in VGPRs, loaded once. Weight slabs stream through a
// double-buffered LDS slab. Scheduling directives prescribe a 12-deep DS-read
// pipeline with a 6-WMMA : 2-DS steady state so the matrix pipe never waits on LDS.
__global__ __launch_bounds__(256, 1) void argmin_kernel(
    const float* __restrict__ z, const float* __restrict__ w,
    const float* __restrict__ esq, int* __restrict__ idx_out) {
  __shared__ float lw[2][SLAB * LDS_STRIDE];

  const int tid   = threadIdx.x;
  const int lane  = tid & 31;
  const int wave  = tid >> 5;
  const int laneN = lane & 15;   // N column / M row selector within half
  const int half  = lane >> 4;   // 0 -> K pair {0,1}, 1 -> K pair {2,3}

  // A fragments per ISA 32-bit 16x4 layout:
  // lane<16: v0=z[M=lane][4s], v1=z[M=lane][4s+1]; lane>=16: K offset +2.
  const int rowBase = blockIdx.x * 128 + wave * 16;
  const float* zrow = z + (size_t)(rowBase + laneN) * D_SZ + half * 2;
  v2f a[NSTEP];
#pragma unroll
  for (int s = 0; s < NSTEP; ++s) a[s] = *(const v2f*)(zrow + s * 4);

  float best[8];
  int   bidx[8];
#pragma unroll
  for (int j = 0; j < 8; ++j) { best[j] = 3.4e38f; bidx[j] = 0; }

  // Per-thread staging decomposition: 16*256 floats / slab = 1024 float4, 4 per thread.
  const int sc = tid >> 6;              // code row within slab (64 float4-chunks per row)
  const int se = (tid & 63) << 2;       // element offset within row
  const int ldsDst = sc * LDS_STRIDE + se;

  // Preload slab 0 into buffer 0.
  {
    const v4f* src = (const v4f*)w;
#pragma unroll
    for (int i = 0; i < 4; ++i) {
      *(v4f*)&lw[0][ldsDst + i * 4 * LDS_STRIDE] = src[tid + i * 256];
    }
  }
  __syncthreads();

  int p = 0;
#pragma unroll 1
  for (int slab = 0; slab < K_SZ; slab += SLAB) {
    // Issue global loads for the NEXT slab first; they drain under the WMMA chain.
    v4f stage[4];
    const bool more = (slab + SLAB) < K_SZ;
    if (more) {
      const v4f* src = (const v4f*)(w + (size_t)(slab + SLAB) * D_SZ);
#pragma unroll
      for (int i = 0; i < 4; ++i) stage[i] = src[tid + i * 256];
    }

    const int n   = slab + laneN;       // global code id this lane owns
    const float e = esq[n];             // ||e_n||^2 (issued early, overlaps WMMAs)
    const float* bp = &lw[p][laneN * LDS_STRIDE + half * 2];

    // Whole tile's B fragments (32x ds_load_2addr_b64 after merging).
    v2f b[NSTEP];
#pragma unroll
    for (int i = 0; i < NSTEP; ++i) b[i] = *(const v2f*)(bp + i * 4);

#if !__has_builtin(__builtin_amdgcn_sched_group_barrier)
#if __has_builtin(__builtin_amdgcn_sched_barrier)
    __builtin_amdgcn_sched_barrier(0);   // hard split: all DS reads before all WMMAs
#else
    asm volatile("" ::: "memory");
#endif
#endif

    v8f c0 = {0.f, 0.f, 0.f, 0.f, 0.f, 0.f, 0.f, 0.f};
    v8f c1 = {0.f, 0.f, 0.f, 0.f, 0.f, 0.f, 0.f, 0.f};
#pragma unroll
    for (int s = 0; s < NSTEP; s += 2) {
      c0 = __builtin_amdgcn_wmma_f32_16x16x4_f32(
               false, a[s],     false, b[s],     (short)0, c0, false, false);
      c1 = __builtin_amdgcn_wmma_f32_16x16x4_f32(
               false, a[s + 1], false, b[s + 1], (short)0, c1, false, false);
    }

#if __has_builtin(__builtin_amdgcn_sched_group_barrier)
    // Prescribed pipeline for this tile (0x100 = DS read, 0x008 = MFMA/WMMA):
    // 12 DS reads in flight up front, then 6 WMMA : 2 DS steady state, tail 4 WMMA.
    __builtin_amdgcn_sched_group_barrier(0x100, 12, 0);
#pragma unroll
    for (int g = 0; g < 10; ++g) {
      __builtin_amdgcn_sched_group_barrier(0x008, 6, 0);
      __builtin_amdgcn_sched_group_barrier(0x100, 2, 0);
    }
    __builtin_amdgcn_sched_group_barrier(0x008, 4, 0);
#endif

    // score = ||e||^2 - 2*cross  (||z||^2 is row-constant, irrelevant for argmin)
#pragma unroll
    for (int j = 0; j < 8; ++j) {
      float sc2 = e - 2.0f * (c0[j] + c1[j]);
      if (sc2 < best[j]) { best[j] = sc2; bidx[j] = n; }
    }

    // Fill the other LDS buffer with the staged slab, then one barrier per slab.
    if (more) {
#pragma unroll
      for (int i = 0; i < 4; ++i) {
        *(v4f*)&lw[p ^ 1][ldsDst + i * 4 * LDS_STRIDE] = stage[i];
      }
    }
    __syncthreads();
    p ^= 1;
  }

  // Cross-lane argmin over the 16 lanes of each half (xor masks < 16 stay in-half).
#pragma unroll
  for (int m = 1; m <= 8; m <<= 1) {
#pragma unroll
    for (int j = 0; j < 8; ++j) {
      float ob = __shfl_xor(best[j], m, 32);
      int   oi = __shfl_xor(bidx[j], m, 32);
      if (ob < best[j] || (ob == best[j] && oi < bidx[j])) {
        best[j] = ob; bidx[j] = oi;
      }
    }
  }
  // VGPR j holds M=j (lanes 0-15) and M=j+8 (lanes 16-31); lane 0 / lane 16 hold results.
  if (laneN == 0) {
    int r = rowBase + half * 8;
#pragma unroll
    for (int j = 0; j < 8; ++j) idx_out[r + j] = bidx[j];
  }
}

// ---------------- Kernel 3: gather z_q_x = weight[idx] ; sg = z + (zq - z) ----------------
__global__ void gather_kernel(const float* __restrict__ z, const float* __restrict__ w,
                              const int* __restrict__ idx,
                              float* __restrict__ sg, float* __restrict__ zq) {
  const int lane = threadIdx.x & 31;
  const int wave = threadIdx.x >> 5;
  const int row  = blockIdx.x * 8 + wave;
  const int k    = idx[row];
  const v4f* ws = (const v4f*)(w + (size_t)k * D_SZ);
  const v4f* zs = (const v4f*)(z + (size_t)row * D_SZ);
  v4f* sgp = (v4f*)(sg + (size_t)row * D_SZ);
  v4f* zqp = (v4f*)(zq + (size_t)row * D_SZ);
#pragma unroll
  for (int i = 0; i < (D_SZ / 4) / 32; ++i) {
    int e = lane + i * 32;
    v4f wv = ws[e];
    v4f zv = zs[e];
    zqp[e] = wv;
    sgp[e] = zv + (wv - zv);   // straight-through estimator, fp32 rounding as in reference
  }
}

extern "C" void kernel_launch(void* const* d_in, const int* in_sizes, int n_in,
                              void* d_out, int out_size, void* d_ws, size_t ws_size,
                              hipStream_t stream) {
  const float* z = (const float*)d_in[0];   // [B, D] fp32
  const float* w = (const float*)d_in[1];   // [K, D] fp32

  float* out = (float*)d_out;
  float* sg  = out;                                   // [B, D]
  float* zq  = out + (size_t)B_SZ * D_SZ;             // [B, D]
  int*   idx = (int*)(out + 2 * (size_t)B_SZ * D_SZ); // [B] int32

  float* esq = (float*)d_ws;                          // [K] scratch

  esq_kernel<<<K_SZ / 8, 256, 0, stream>>>(w, esq);
  argmin_kernel<<<B_SZ / 128, 256, 0, stream>>>(z, w, esq, idx);
  gather_kernel<<<B_SZ / 8, 256, 0, stream>>>(z, w, idx, sg, zq);
}